// SortingRegressionModel_38903813767817
// MI455X (gfx1250) — compile-verified
//
#include <hip/hip_runtime.h>
#include <hip/hip_bf16.h>
#include <stdint.h>

// ---------------------------------------------------------------------------
// Soft bitonic sort (N = 2^21) with fused head/tail linears for MI455X.
//  - 8 MB working vector lives in d_out and stays L2-resident (192 MB L2).
//  - 32 KB LDS tiles: all substages with j <= TILE/2 run in LDS.
//  - Tile load AND tile stores use the CDNA5 Tensor Data Mover
//    (tensor_load_to_lds / tensor_store_from_lds + s_wait_tensorcnt).
//  - Global substages (j >= TILE) are 4x-vectorized b128 L2 streaming passes.
// ---------------------------------------------------------------------------

#define N_ELEMS  (1 << 21)     // 2097152
#define TILE     8192          // elements per LDS tile (32 KB fp32)
#define NTHREADS 256           // 8 wave32 per workgroup
#define NPAIRS   (TILE / 2)

#ifndef __has_builtin
#define __has_builtin(x) 0
#endif

#if __has_builtin(__builtin_amdgcn_tensor_load_to_lds) && \
    __has_builtin(__builtin_amdgcn_tensor_store_from_lds) && \
    __has_builtin(__builtin_amdgcn_s_wait_tensorcnt)
#define USE_TDM 1
#else
#define USE_TDM 0
#endif

typedef unsigned int u32x4 __attribute__((ext_vector_type(4)));
typedef int          i32x4 __attribute__((ext_vector_type(4)));
typedef int          i32x8 __attribute__((ext_vector_type(8)));

// Soft compare-exchange:  s = sigmoid(b - a); mn = s*a + (1-s)*b; mx = a+b-mn
__device__ __forceinline__ void soft_cex(float a, float b, float& mn, float& mx) {
  float t = __expf(a - b);          // exp(a-b)
  float s = 1.0f / (1.0f + t);      // sigmoid(b-a)
  mn = fmaf(s, a - b, b);           // s*a + (1-s)*b   (symmetric in a,b)
  mx = (a + b) - mn;
}

#if USE_TDM
// 1-D TDM descriptor: one contiguous tile of TILE fp32 at `ga` <-> LDS `lds_off`.
__device__ __forceinline__ void tdm_desc(uint32_t lds_off, uint64_t ga,
                                         u32x4& g0, i32x8& g1) {
  g0[0] = 1u;                                         // count=1 (valid)
  g0[1] = lds_off;                                    // LDS byte offset
  g0[2] = (uint32_t)ga;                               // global_addr[31:0]
  g0[3] = ((uint32_t)(ga >> 32) & 0x01FFFFFFu) | (2u << 30); // [56:32] | type=2

  g1[0] = 0x00020000;                                 // wg_mask=0, data_size=2 (4B)
  g1[1] = (int)(((unsigned)TILE & 0xFFFFu) << 16);    // tensor_dim0[15:0]
  g1[2] = (int)(((unsigned)TILE >> 16) | (1u << 16)); // dim0 hi | tensor_dim1=1 lo
  g1[3] = (int)((unsigned)TILE << 16);                // tensor_dim1 hi | tile_dim0
  g1[4] = 0;                                          // tile_dim1/2 unused
  g1[5] = TILE;                                       // tensor_dim0_stride[31:0]
  g1[6] = 0;
  g1[7] = 0;
}

__device__ __forceinline__ void tdm_load_tile(void* lds_ptr, const float* gptr) {
  u32x4 g0; i32x8 g1;
  tdm_desc((uint32_t)(uintptr_t)lds_ptr, (uint64_t)(uintptr_t)gptr, g0, g1);
  i32x4 gz = {0, 0, 0, 0};
#if __clang_major__ >= 23
  i32x8 gz8 = {0, 0, 0, 0, 0, 0, 0, 0};
  __builtin_amdgcn_tensor_load_to_lds(g0, g1, gz, gz, gz8, 0);
#else
  __builtin_amdgcn_tensor_load_to_lds(g0, g1, gz, gz, 0);
#endif
  __builtin_amdgcn_s_wait_tensorcnt(0);
}

__device__ __forceinline__ void tdm_store_tile(void* lds_ptr, float* gptr) {
  u32x4 g0; i32x8 g1;
  tdm_desc((uint32_t)(uintptr_t)lds_ptr, (uint64_t)(uintptr_t)gptr, g0, g1);
  i32x4 gz = {0, 0, 0, 0};
#if __clang_major__ >= 23
  i32x8 gz8 = {0, 0, 0, 0, 0, 0, 0, 0};
  __builtin_amdgcn_tensor_store_from_lds(g0, g1, gz, gz, gz8, 0);
#else
  __builtin_amdgcn_tensor_store_from_lds(g0, g1, gz, gz, 0);
#endif
  // TENSORcnt drained by S_ENDPGM's implicit wait-idle; LDS lives until the
  // whole workgroup retires, so the DMA read cannot be orphaned.
}
#endif  // USE_TDM

// ---------------------------------------------------------------------------
// Kernel A: h = x@w1 + b1 (vectorized), then stages k = 2 .. TILE in LDS,
// tile store via TDM.
// ---------------------------------------------------------------------------
__global__ void __launch_bounds__(NTHREADS)
kern_head_sort(const float* __restrict__ x,
               const float* __restrict__ w1,
               const float* __restrict__ b1,
               float* __restrict__ buf) {
  __shared__ float lds[TILE];
  const int base = blockIdx.x * TILE;
  const float c0 = w1[0], c1 = w1[1], c2 = w1[2], bb = b1[0];

  // 4 elements per iteration: 3 aligned float4 loads of the [N,3] rows.
  for (int e4 = threadIdx.x; e4 < TILE / 4; e4 += NTHREADS) {
    const int g0e = base + 4 * e4;
    const float4* xv = (const float4*)(x + 3ull * (unsigned)g0e);
    float4 v0 = xv[0], v1 = xv[1], v2 = xv[2];
    float4 h;
    h.x = fmaf(v0.x, c0, fmaf(v0.y, c1, fmaf(v0.z, c2, bb)));
    h.y = fmaf(v0.w, c0, fmaf(v1.x, c1, fmaf(v1.y, c2, bb)));
    h.z = fmaf(v1.z, c0, fmaf(v1.w, c1, fmaf(v2.x, c2, bb)));
    h.w = fmaf(v2.y, c0, fmaf(v2.z, c1, fmaf(v2.w, c2, bb)));
    ((float4*)lds)[e4] = h;
  }
  __syncthreads();

  for (int k = 2; k <= TILE; k <<= 1) {
    for (int j = k >> 1; j >= 1; j >>= 1) {
      const int jm1 = j - 1;
      for (int q = threadIdx.x; q < NPAIRS; q += NTHREADS) {
        const int i = ((q & ~jm1) << 1) | (q & jm1);  // low wire (bit j clear)
        const int p = i + j;                          // high wire
        float a = lds[i], b = lds[p], mn, mx;
        soft_cex(a, b, mn, mx);
        const bool asc = (((base + i) & k) == 0);
        lds[i] = asc ? mn : mx;
        lds[p] = asc ? mx : mn;
      }
      __syncthreads();
    }
  }

#if USE_TDM
  if (threadIdx.x < 32) tdm_store_tile(lds, buf + base);
#else
  for (int e4 = threadIdx.x; e4 < TILE / 4; e4 += NTHREADS)
    ((float4*)(buf + base))[e4] = ((const float4*)lds)[e4];
#endif
}

// ---------------------------------------------------------------------------
// Kernel C: one global substage (k, j), j >= TILE. 4 pairs/thread, b128 I/O.
// ---------------------------------------------------------------------------
__global__ void __launch_bounds__(NTHREADS)
kern_global_cex4(float* __restrict__ buf, int k, int j) {
  const int q   = (blockIdx.x * NTHREADS + threadIdx.x) << 2;  // 4 pair ids
  const int jm1 = j - 1;
  const int i   = ((q & ~jm1) << 1) | (q & jm1);  // 4 consecutive low wires
  const int p   = i + j;
  float4 a = *(const float4*)(buf + i);
  float4 b = *(const float4*)(buf + p);
  float4 mn, mx;
  soft_cex(a.x, b.x, mn.x, mx.x);
  soft_cex(a.y, b.y, mn.y, mx.y);
  soft_cex(a.z, b.z, mn.z, mx.z);
  soft_cex(a.w, b.w, mn.w, mx.w);
  const bool asc = ((i & k) == 0);  // uniform across the 4 (k >= 2*TILE)
  *(float4*)(buf + i) = asc ? mn : mx;
  *(float4*)(buf + p) = asc ? mx : mn;
}

// ---------------------------------------------------------------------------
// Kernel B: tail of stage k — substages j = TILE/2 .. 1 in LDS.
// TDM tile load; TDM tile store (or fused out = s*w2 + b2 when fin).
// ---------------------------------------------------------------------------
__global__ void __launch_bounds__(NTHREADS)
kern_tile_tail(float* __restrict__ buf, int k, int fin,
               const float* __restrict__ w2,
               const float* __restrict__ b2) {
  __shared__ float lds[TILE];
  const int base = blockIdx.x * TILE;

#if USE_TDM
  if (threadIdx.x < 32) tdm_load_tile(lds, buf + base);  // wave 0 issues DMA
#else
  for (int e4 = threadIdx.x; e4 < TILE / 4; e4 += NTHREADS)
    ((float4*)lds)[e4] = ((const float4*)(buf + base))[e4];
#endif
  __syncthreads();

  const bool asc = ((base & k) == 0);  // block-uniform for k >= 2*TILE
  for (int j = TILE >> 1; j >= 1; j >>= 1) {
    const int jm1 = j - 1;
    for (int q = threadIdx.x; q < NPAIRS; q += NTHREADS) {
      const int i = ((q & ~jm1) << 1) | (q & jm1);
      const int p = i + j;
      float a = lds[i], b = lds[p], mn, mx;
      soft_cex(a, b, mn, mx);
      lds[i] = asc ? mn : mx;
      lds[p] = asc ? mx : mn;
    }
    __syncthreads();
  }

  if (fin) {
    const float w = w2[0], bb = b2[0];
    for (int e4 = threadIdx.x; e4 < TILE / 4; e4 += NTHREADS) {
      float4 v = ((const float4*)lds)[e4];
      float4 o;
      o.x = fmaf(v.x, w, bb);
      o.y = fmaf(v.y, w, bb);
      o.z = fmaf(v.z, w, bb);
      o.w = fmaf(v.w, w, bb);
      ((float4*)(buf + base))[e4] = o;
    }
  } else {
#if USE_TDM
    if (threadIdx.x < 32) tdm_store_tile(lds, buf + base);
#else
    for (int e4 = threadIdx.x; e4 < TILE / 4; e4 += NTHREADS)
      ((float4*)(buf + base))[e4] = ((const float4*)lds)[e4];
#endif
  }
}

// ---------------------------------------------------------------------------
// Host-side launch sequence (graph-capture safe: kernels only, all on stream).
// d_out doubles as the 8 MB working vector (fully overwritten every call).
// ---------------------------------------------------------------------------
extern "C" void kernel_launch(void* const* d_in, const int* in_sizes, int n_in,
                              void* d_out, int out_size, void* d_ws, size_t ws_size,
                              hipStream_t stream) {
  (void)in_sizes; (void)n_in; (void)out_size; (void)d_ws; (void)ws_size;
  const float* x  = (const float*)d_in[0];
  const float* w1 = (const float*)d_in[1];
  const float* b1 = (const float*)d_in[2];
  const float* w2 = (const float*)d_in[3];
  const float* b2 = (const float*)d_in[4];
  float* buf = (float*)d_out;

  const int ntiles  = N_ELEMS / TILE;                  // 256 workgroups
  const int npairb4 = (N_ELEMS / 2) / 4 / NTHREADS;    // 1024 workgroups

  // linear_1 + stages k = 2 .. TILE (91 substages) fused in LDS
  kern_head_sort<<<ntiles, NTHREADS, 0, stream>>>(x, w1, b1, buf);

  // stages k = 2*TILE .. N
  for (int k = TILE * 2; k <= N_ELEMS; k <<= 1) {
    for (int j = k >> 1; j >= TILE; j >>= 1)   // global substages (36 total)
      kern_global_cex4<<<npairb4, NTHREADS, 0, stream>>>(buf, k, j);
    // substages j = TILE/2 .. 1 in LDS; last stage fuses linear_2
    kern_tile_tail<<<ntiles, NTHREADS, 0, stream>>>(
        buf, k, (k == N_ELEMS) ? 1 : 0, w2, b2);
  }
}